// Encoder_81192061763799
// MI455X (gfx1250) — compile-verified
//
#include <hip/hip_runtime.h>
#include <hip/hip_bf16.h>
#include <math.h>

// ---------------------------------------------------------------------------
// Shapes (fixed by the reference): S=256, B=64, IN=3, H=128, P_IN=8, P_OUT=32,
// NUM_GRAPHS=64. N (nodes) taken from in_sizes.
// d_out layout: outputs [S,B,H] (2,097,152 f32) | hidden [B,H] (8192 f32) | lg [N] f32
// ---------------------------------------------------------------------------

#define SS 256
#define BB 64
#define HH 128
#define NGRAPH 64
#define EPSV 1e-6f

typedef float v2f __attribute__((ext_vector_type(2)));
typedef float v8f __attribute__((ext_vector_type(8)));

// float atomic-max via integer punning (valid for LDS and global pointers).
template <typename P>
__device__ inline void atomic_max_float_pun(P* addr, float val) {
    if (!(val == val)) return;                  // ignore NaN
    if (val >= 0.0f) atomicMax((int*)addr, __float_as_int(val));
    else             atomicMin((unsigned int*)addr, __float_as_uint(val));
}

// s_wait_asynccnt 0 — wait for all outstanding async global<->LDS transfers.
__device__ inline void wait_asynccnt0() {
#if __has_builtin(__builtin_amdgcn_s_wait_asynccnt)
    __builtin_amdgcn_s_wait_asynccnt(0);
#else
    asm volatile("s_wait_asynccnt 0x0" ::: "memory");
#endif
}

// Direct global -> LDS async copy of 16 bytes per lane (GLOBAL_LOAD_ASYNC_TO_LDS_B128,
// GV addressing; ASYNCcnt-tracked). ldsa = 32-bit LDS byte address, ga = 64-bit
// global address. Per-lane EXEC predication applies.
__device__ inline void async_g2lds_b128(unsigned int ldsa, unsigned long long ga) {
    asm volatile("global_load_async_to_lds_b128 %0, %1, off"
                 :: "v"(ldsa), "v"(ga)
                 : "memory");
}

// ---------------------------------------------------------------------------
// K0: init per-graph stats in workspace
// ---------------------------------------------------------------------------
__global__ void k_init_stats(float* gmax, float* gsum) {
    int t = threadIdx.x;
    if (t < NGRAPH) { gmax[t] = -INFINITY; gsum[t] = 0.0f; }
}

// ---------------------------------------------------------------------------
// K1: outputs[s,b,h] = (src[s,b,:]@W_in[:,h] + b_in[h]) * (s < len[b])
// ---------------------------------------------------------------------------
__global__ __launch_bounds__(256) void k_encoder_out(
    const float* __restrict__ src, const int* __restrict__ src_len,
    const float* __restrict__ W_in, const float* __restrict__ b_in,
    float* __restrict__ out) {
    int idx = blockIdx.x * 256 + threadIdx.x;
    if (idx >= SS * BB * HH) return;
    int h = idx & (HH - 1);
    int b = (idx >> 7) & (BB - 1);
    int s = idx >> 13;
    const float* p = src + (s * BB + b) * 3;
    float v = p[0] * W_in[h] + p[1] * W_in[HH + h] + p[2] * W_in[2 * HH + h] + b_in[h];
    out[idx] = (s < src_len[b]) ? v : 0.0f;
}

// ---------------------------------------------------------------------------
// K2: hidden = tanh(concat(tanh(pro@W_extra+b_extra), mean_s(outputs)) @ W_hid + b_hid)
// mean computed analytically: (sum_{s<len} src[s,b,:]) @ W_in + len*b_in, / S
// One block per batch element, 128 threads (one per h).
// ---------------------------------------------------------------------------
__global__ __launch_bounds__(128) void k_encoder_hidden(
    const float* __restrict__ src, const int* __restrict__ src_len,
    const float* __restrict__ pro,
    const float* __restrict__ W_in, const float* __restrict__ b_in,
    const float* __restrict__ W_extra, const float* __restrict__ b_extra,
    const float* __restrict__ W_hid, const float* __restrict__ b_hid,
    float* __restrict__ out_hidden) {
    int b = blockIdx.x;
    int t = threadIdx.x;
    __shared__ float red[3][128];
    __shared__ float cat[32 + HH];

    int len = src_len[b];
    float s0 = 0.f, s1 = 0.f, s2 = 0.f;
    for (int s = t; s < SS; s += 128) {
        if (s < len) {
            const float* p = src + (s * BB + b) * 3;
            s0 += p[0]; s1 += p[1]; s2 += p[2];
        }
    }
    red[0][t] = s0; red[1][t] = s1; red[2][t] = s2;
    __syncthreads();
    for (int off = 64; off > 0; off >>= 1) {
        if (t < off) {
            red[0][t] += red[0][t + off];
            red[1][t] += red[1][t + off];
            red[2][t] += red[2][t + off];
        }
        __syncthreads();
    }
    float c0 = red[0][0], c1 = red[1][0], c2 = red[2][0];

    float hp = (c0 * W_in[t] + c1 * W_in[HH + t] + c2 * W_in[2 * HH + t] +
                (float)len * b_in[t]) * (1.0f / (float)SS);
    cat[32 + t] = hp;
    if (t < 32) {
        float acc = b_extra[t];
        #pragma unroll
        for (int q = 0; q < 8; ++q) acc += pro[b * 8 + q] * W_extra[q * 32 + t];
        cat[t] = tanhf(acc);
    }
    __syncthreads();

    float acc = b_hid[t];
    for (int k = 0; k < 32 + HH; ++k) acc += cat[k] * W_hid[k * HH + t];
    out_hidden[b * HH + t] = tanhf(acc);
}

// ---------------------------------------------------------------------------
// K3: lg[n] = node_x[n,:] @ W_pred + b_pred   via V_WMMA_F32_16X16X4_F32,
// staged through LDS with GLOBAL_LOAD_ASYNC_TO_LDS_B128 (ASYNCcnt path),
// plus per-graph running max (LDS -> global punned atomics).
// Block = 128 threads = 4 waves; block tile = 64 rows of node_x in LDS.
// A layout (ISA 16x4 f32): lanes 0-15 row m k={0,1}, lanes 16-31 row m k={2,3}.
// B is the 4-long W_pred chunk replicated across all 16 columns, so column 0
// of D holds the dot products; lanes 0 and 16 extract it (C/D f32 layout).
// ---------------------------------------------------------------------------
#define TILE_M 64
#define LDS_STRIDE 132   // 128 + 4: float4-aligned, rotates LDS banks per row

__global__ __launch_bounds__(128) void k_lg_wmma(
    const float* __restrict__ node_x, const int* __restrict__ gid,
    const float* __restrict__ W_pred, const float* __restrict__ b_pred,
    int N, float* __restrict__ lg, float* __restrict__ gmax) {
    __shared__ float tileA[TILE_M * LDS_STRIDE];
    __shared__ float wp[HH];
    __shared__ float smax[NGRAPH];

    int t = threadIdx.x;
    if (t < NGRAPH) smax[t] = -INFINITY;
    wp[t] = W_pred[t];                      // 128 threads, 128 weights
    int tile0 = blockIdx.x * TILE_M;

    // Coalesced async stage: 64 rows x 128 cols, 16 bytes/lane straight to LDS
    // (2048 float4, 16 per thread). Tail lanes zero-fill LDS instead.
    #pragma unroll
    for (int i = 0; i < 16; ++i) {
        int e  = t + i * 128;               // 0..2047
        int r  = e >> 5;                    // 32 float4 per row
        int c4 = e & 31;
        float* ldst = &tileA[r * LDS_STRIDE + c4 * 4];
        if (tile0 + r < N) {
            unsigned int ldsa = (unsigned int)(uintptr_t)(void*)ldst;
            unsigned long long ga =
                (unsigned long long)(uintptr_t)(node_x + (size_t)(tile0 + r) * HH + c4 * 4);
            async_g2lds_b128(ldsa, ga);
        } else {
            *reinterpret_cast<float4*>(ldst) = make_float4(0.f, 0.f, 0.f, 0.f);
        }
    }
    wait_asynccnt0();
    __syncthreads();

    int lane  = t & 31;
    int wv    = t >> 5;                     // wave 0..3, rows [wv*16, wv*16+16)
    int rlo   = lane & 15;
    int khalf = (lane >> 4) << 1;           // 0 for lanes 0-15, 2 for 16-31
    const float* arow = &tileA[(wv * 16 + rlo) * LDS_STRIDE + khalf];

    v8f acc = {};
    #pragma unroll
    for (int kk = 0; kk < HH / 4; ++kk) {
        v2f a;  a.x  = arow[kk * 4];          a.y  = arow[kk * 4 + 1];
        v2f bb; bb.x = wp[kk * 4 + khalf];    bb.y = wp[kk * 4 + khalf + 1];
        acc = __builtin_amdgcn_wmma_f32_16x16x4_f32(
            false, a, false, bb, (short)0, acc, false, false);
    }

    float bp = b_pred[0];
    // Column 0 of D: lane 0 holds M=0..7 in acc[0..7], lane 16 holds M=8..15.
    if (lane == 0 || lane == 16) {
        int mbase = wv * 16 + ((lane >> 4) << 3);
        #pragma unroll
        for (int i = 0; i < 8; ++i) {
            int row = tile0 + mbase + i;
            if (row < N) {
                float v = acc[i] + bp;
                lg[row] = v;
                atomic_max_float_pun(&smax[gid[row]], v);
            }
        }
    }
    __syncthreads();
    if (t < NGRAPH) atomic_max_float_pun(&gmax[t], smax[t]);
}

// ---------------------------------------------------------------------------
// K4: e = exp(lg - max[g]) * w   (in place), per-graph sums via LDS -> global
// ---------------------------------------------------------------------------
__global__ __launch_bounds__(256) void k_seg_expsum(
    float* __restrict__ lg, const float* __restrict__ node_w,
    const int* __restrict__ gid, const float* __restrict__ gmax,
    float* __restrict__ gsum, int N) {
    __shared__ float ssum[NGRAPH];
    int t = threadIdx.x;
    if (t < NGRAPH) ssum[t] = 0.0f;
    __syncthreads();
    for (int n = blockIdx.x * 256 + t; n < N; n += gridDim.x * 256) {
        int g = gid[n];
        float e = expf(lg[n] - gmax[g]) * node_w[n];
        lg[n] = e;
        atomicAdd(&ssum[g], e);
    }
    __syncthreads();
    if (t < NGRAPH) atomicAdd(&gsum[t], ssum[t]);
}

// ---------------------------------------------------------------------------
// K5: out = log(clip(e / (sum[g] + eps), eps, 1))   (in place)
// ---------------------------------------------------------------------------
__global__ __launch_bounds__(256) void k_seg_finalize(
    float* __restrict__ lg, const int* __restrict__ gid,
    const float* __restrict__ gsum, int N) {
    int n = blockIdx.x * 256 + threadIdx.x;
    if (n < N) {
        float pred = lg[n] / (gsum[gid[n]] + EPSV);
        pred = fminf(fmaxf(pred, EPSV), 1.0f);
        lg[n] = logf(pred);
    }
}

// ---------------------------------------------------------------------------
extern "C" void kernel_launch(void* const* d_in, const int* in_sizes, int n_in,
                              void* d_out, int out_size, void* d_ws, size_t ws_size,
                              hipStream_t stream) {
    const float* src      = (const float*)d_in[0];
    const int*   src_len  = (const int*)  d_in[1];
    const float* node_x   = (const float*)d_in[2];
    const float* node_w   = (const float*)d_in[3];
    const int*   gid      = (const int*)  d_in[4];
    const float* pro      = (const float*)d_in[5];
    const float* W_in     = (const float*)d_in[6];
    const float* b_in     = (const float*)d_in[7];
    const float* W_pred   = (const float*)d_in[8];
    const float* b_pred   = (const float*)d_in[9];
    const float* W_extra  = (const float*)d_in[10];
    const float* b_extra  = (const float*)d_in[11];
    const float* W_hid    = (const float*)d_in[12];
    const float* b_hid    = (const float*)d_in[13];

    const int N = in_sizes[3];                       // node count (node_w)

    float* out        = (float*)d_out;
    float* out_outs   = out;                         // [S,B,H]
    float* out_hidden = out + SS * BB * HH;          // [B,H]
    float* out_lg     = out_hidden + BB * HH;        // [N]

    float* gmax = (float*)d_ws;                      // 64 f32
    float* gsum = gmax + NGRAPH;                     // 64 f32

    k_init_stats<<<1, 64, 0, stream>>>(gmax, gsum);

    k_encoder_out<<<(SS * BB * HH) / 256, 256, 0, stream>>>(
        src, src_len, W_in, b_in, out_outs);

    k_encoder_hidden<<<BB, HH, 0, stream>>>(
        src, src_len, pro, W_in, b_in, W_extra, b_extra, W_hid, b_hid, out_hidden);

    k_lg_wmma<<<(N + TILE_M - 1) / TILE_M, 128, 0, stream>>>(
        node_x, gid, W_pred, b_pred, N, out_lg, gmax);

    int blocks4 = (N + 255) / 256; if (blocks4 > 1024) blocks4 = 1024;
    k_seg_expsum<<<blocks4, 256, 0, stream>>>(out_lg, node_w, gid, gmax, gsum, N);

    k_seg_finalize<<<(N + 255) / 256, 256, 0, stream>>>(out_lg, gid, gsum, N);
}